// SAGE_53781580480527
// MI455X (gfx1250) — compile-verified
//
#include <hip/hip_runtime.h>
#include <hip/hip_bf16.h>

typedef __attribute__((ext_vector_type(2))) float v2f;
typedef __attribute__((ext_vector_type(8))) float v8f;

#define EPS_BN   1e-5f
#define EPS_NORM 1e-12f

// -------- fp32 WMMA wrapper: D = A(16x4) * B(4x16) + C, exact f32 math --------
__device__ __forceinline__ v8f wmma_f32(v2f a, v2f b, v8f c) {
  // 8 args: (neg_a, A, neg_b, B, c_mod, C, reuse_a, reuse_b)
  return __builtin_amdgcn_wmma_f32_16x16x4_f32(false, a, false, b, (short)0, c,
                                               false, false);
}

// ============================ BatchNorm =====================================
// Pass 1: per-block partial sums of x and x^2 (deterministic tree reduce).
__global__ void bn_partial_k(const float* __restrict__ h, float* __restrict__ part, int n) {
  __shared__ float sh[256 * 10];
  float acc[10];
#pragma unroll
  for (int d = 0; d < 10; ++d) acc[d] = 0.f;
  for (int i = blockIdx.x * blockDim.x + threadIdx.x; i < n;
       i += gridDim.x * blockDim.x) {
#pragma unroll
    for (int d = 0; d < 5; ++d) {
      float v = h[i * 5 + d];
      acc[d] += v;
      acc[5 + d] += v * v;
    }
  }
#pragma unroll
  for (int d = 0; d < 10; ++d) sh[threadIdx.x * 10 + d] = acc[d];
  __syncthreads();
  for (int s = 128; s > 0; s >>= 1) {
    if (threadIdx.x < (unsigned)s)
      for (int d = 0; d < 10; ++d)
        sh[threadIdx.x * 10 + d] += sh[(threadIdx.x + s) * 10 + d];
    __syncthreads();
  }
  if (threadIdx.x < 10) part[blockIdx.x * 10 + threadIdx.x] = sh[threadIdx.x];
}

// Pass 2: fold partials -> mu, rsqrt(var+eps). Deterministic serial fold.
__global__ void bn_final_k(const float* __restrict__ part, float* __restrict__ mu,
                           float* __restrict__ rs, int nblocks, float inv_n) {
  int t = threadIdx.x;
  if (t < 5) {
    float s = 0.f, q = 0.f;
    for (int b = 0; b < nblocks; ++b) {
      s += part[b * 10 + t];
      q += part[b * 10 + 5 + t];
    }
    float m = s * inv_n;
    float var = q * inv_n - m * m;
    mu[t] = m;
    rs[t] = rsqrtf(var + EPS_BN);
  }
}

__global__ void bn_apply_k(const float* __restrict__ h, const float* __restrict__ mu,
                           const float* __restrict__ rs, const float* __restrict__ g,
                           const float* __restrict__ b, float* __restrict__ x, int total) {
  int i = blockIdx.x * blockDim.x + threadIdx.x;
  if (i >= total) return;
  int d = i % 5;
  x[i] = (h[i] - mu[d]) * rs[d] * g[d] + b[d];
}

// ============================ Degrees =======================================
__global__ void degree_k(const int* __restrict__ dst, float* __restrict__ cnt, int E) {
  int e = blockIdx.x * blockDim.x + threadIdx.x;
  if (e < E) atomicAdd(&cnt[dst[e]], 1.0f);
}

__global__ void invcnt_k(const float* __restrict__ cnt, float* __restrict__ invc, int n) {
  int i = blockIdx.x * blockDim.x + threadIdx.x;
  if (i < n) invc[i] = 1.0f / fmaxf(cnt[i], 1.0f);
}

// ======================= Layer 1 (dim-5) scalar path ========================
__global__ void proj5_k(const float* __restrict__ X, const float* __restrict__ W,
                        const float* __restrict__ b, float* __restrict__ Y, int n) {
  int i = blockIdx.x * blockDim.x + threadIdx.x;
  if (i >= n) return;
  float x[5];
#pragma unroll
  for (int k = 0; k < 5; ++k) x[k] = X[i * 5 + k];
#pragma unroll
  for (int c = 0; c < 5; ++c) {
    float acc = b[c];
#pragma unroll
    for (int k = 0; k < 5; ++k) acc += x[k] * W[k * 5 + c];
    Y[i * 5 + c] = fmaxf(acc, 0.f);
  }
}

__global__ void edge_scatter5_k(const int* __restrict__ src, const int* __restrict__ dst,
                                const float* __restrict__ ew, const float* __restrict__ xp,
                                float* __restrict__ agg, int E) {
  int e = blockIdx.x * blockDim.x + threadIdx.x;
  if (e >= E) return;
  int s = src[e], d = dst[e];
  float w = ew[e];
#pragma unroll
  for (int k = 0; k < 5; ++k) atomicAdd(&agg[d * 5 + k], xp[s * 5 + k] * w);
}

// out = mean(5) @ LW(5x64) + lb + x(5) @ RW(5x64); L2-normalize row; ReLU.
// One 64-thread block per node (2 waves).
__global__ void node5to64_k(const float* __restrict__ agg, const float* __restrict__ invc,
                            const float* __restrict__ X, const float* __restrict__ LW,
                            const float* __restrict__ lb, const float* __restrict__ RW,
                            float* __restrict__ Y, int n) {
  __shared__ float red[64];
  int node = blockIdx.x;
  if (node >= n) return;
  int c = threadIdx.x;
  float ic = invc[node];
  float acc = lb[c];
#pragma unroll
  for (int k = 0; k < 5; ++k) {
    acc += agg[node * 5 + k] * ic * LW[k * 64 + c] + X[node * 5 + k] * RW[k * 64 + c];
  }
  red[c] = acc * acc;
  __syncthreads();
#pragma unroll
  for (int s = 32; s > 0; s >>= 1) {
    if (c < s) red[c] += red[c + s];
    __syncthreads();
  }
  float scale = 1.0f / fmaxf(sqrtf(red[0]), EPS_NORM);
  Y[node * 64 + c] = fmaxf(acc * scale, 0.f);
}

// ======================= Dim-64 WMMA paths ==================================
// Y = relu(X(n x 64) @ W(64x64) + bias). 4 waves/block, 16 rows per block,
// each wave owns one 16x16 output tile; K-loop = 16 chained f32 WMMAs.
__global__ void proj64_wmma_k(const float* __restrict__ X, const float* __restrict__ W,
                              const float* __restrict__ bias, float* __restrict__ Y, int n) {
  int l = threadIdx.x & 31;
  int wave = threadIdx.x >> 5;
  int row0 = blockIdx.x * 16;
  int m = row0 + (l & 15);
  if (m >= n) m = n - 1;               // clamp reads; stores are guarded
  int cl = wave * 16 + (l & 15);
  int khi = (l >> 4) << 1;             // lanes 16-31 take K+2
  v8f c = {};
#pragma unroll
  for (int k0 = 0; k0 < 64; k0 += 4) {
    int kb = k0 + khi;
    v2f a, bb;
    a[0] = X[m * 64 + kb];
    a[1] = X[m * 64 + kb + 1];
    bb[0] = W[kb * 64 + cl];
    bb[1] = W[(kb + 1) * 64 + cl];
    c = wmma_f32(a, bb, c);
  }
  int rbase = row0 + ((l >> 4) << 3);  // lanes 16-31 hold rows M+8..M+15
  float bc = bias[cl];
#pragma unroll
  for (int r = 0; r < 8; ++r) {
    int rr = rbase + r;
    if (rr < n) Y[rr * 64 + cl] = fmaxf(c[r] + bc, 0.f);
  }
}

// 16 threads/edge, 4 channels each: agg[dst] += xp[src] * w  (f32 atomics in L2).
__global__ void edge_scatter64_k(const int* __restrict__ src, const int* __restrict__ dst,
                                 const float* __restrict__ ew, const float* __restrict__ xp,
                                 float* __restrict__ agg, int E) {
  int t = blockIdx.x * blockDim.x + threadIdx.x;
  int e = t >> 4;
  if (e >= E) return;
  int c4 = (t & 15) << 2;
  int s = src[e], d = dst[e];
  float w = ew[e];
  const float4 v = *(const float4*)(xp + (size_t)s * 64 + c4);
  float* out = agg + (size_t)d * 64 + c4;
  atomicAdd(out + 0, v.x * w);
  atomicAdd(out + 1, v.y * w);
  atomicAdd(out + 2, v.z * w);
  atomicAdd(out + 3, v.w * w);
}

// out = (agg*invc) @ LW + lb + X @ RW; L2-normalize rows; ReLU. Fused via LDS tile.
__global__ void node64_wmma_k(const float* __restrict__ agg, const float* __restrict__ invc,
                              const float* __restrict__ X, const float* __restrict__ LW,
                              const float* __restrict__ lb, const float* __restrict__ RW,
                              float* __restrict__ Y, int n) {
  __shared__ float tile[16 * 64];
  __shared__ float rscale[16];
  int l = threadIdx.x & 31;
  int wave = threadIdx.x >> 5;
  int row0 = blockIdx.x * 16;
  int m = row0 + (l & 15);
  if (m >= n) m = n - 1;
  int cl = wave * 16 + (l & 15);
  int khi = (l >> 4) << 1;
  float ic = invc[m];
  v8f c = {};
#pragma unroll
  for (int k0 = 0; k0 < 64; k0 += 4) {      // mean @ LW
    int kb = k0 + khi;
    v2f a, bb;
    a[0] = agg[m * 64 + kb] * ic;
    a[1] = agg[m * 64 + kb + 1] * ic;
    bb[0] = LW[kb * 64 + cl];
    bb[1] = LW[(kb + 1) * 64 + cl];
    c = wmma_f32(a, bb, c);
  }
#pragma unroll
  for (int k0 = 0; k0 < 64; k0 += 4) {      // + x @ RW
    int kb = k0 + khi;
    v2f a, bb;
    a[0] = X[m * 64 + kb];
    a[1] = X[m * 64 + kb + 1];
    bb[0] = RW[kb * 64 + cl];
    bb[1] = RW[(kb + 1) * 64 + cl];
    c = wmma_f32(a, bb, c);
  }
  int rloc = (l >> 4) << 3;
  float bc = lb[cl];
#pragma unroll
  for (int r = 0; r < 8; ++r) tile[(rloc + r) * 64 + cl] = c[r] + bc;
  __syncthreads();
  if (threadIdx.x < 16) {
    float ss = 0.f;
#pragma unroll
    for (int cc = 0; cc < 64; ++cc) {
      float v = tile[threadIdx.x * 64 + cc];
      ss += v * v;
    }
    rscale[threadIdx.x] = 1.0f / fmaxf(sqrtf(ss), EPS_NORM);
  }
  __syncthreads();
  for (int i = threadIdx.x; i < 16 * 64; i += blockDim.x) {
    int r = i >> 6;
    int rr = row0 + r;
    if (rr < n) Y[rr * 64 + (i & 63)] = fmaxf(tile[i] * rscale[r], 0.f);
  }
}

// Final layer: out(n x 5) = mean@LW(64x5) + lb + X@RW(64x5); L2-normalize.
// B padded to 16 cols with zeros; one wave per 16 rows.
__global__ void node64to5_wmma_k(const float* __restrict__ agg, const float* __restrict__ invc,
                                 const float* __restrict__ X, const float* __restrict__ LW,
                                 const float* __restrict__ lb, const float* __restrict__ RW,
                                 float* __restrict__ Y, int n) {
  __shared__ float tile[16 * 16];
  int l = threadIdx.x & 31;
  int row0 = blockIdx.x * 16;
  int m = row0 + (l & 15);
  if (m >= n) m = n - 1;
  int cl = l & 15;
  int khi = (l >> 4) << 1;
  float ic = invc[m];
  bool cv = (cl < 5);
  v8f c = {};
#pragma unroll
  for (int k0 = 0; k0 < 64; k0 += 4) {
    int kb = k0 + khi;
    v2f a, bb;
    a[0] = agg[m * 64 + kb] * ic;
    a[1] = agg[m * 64 + kb + 1] * ic;
    bb[0] = cv ? LW[kb * 5 + cl] : 0.f;
    bb[1] = cv ? LW[(kb + 1) * 5 + cl] : 0.f;
    c = wmma_f32(a, bb, c);
  }
#pragma unroll
  for (int k0 = 0; k0 < 64; k0 += 4) {
    int kb = k0 + khi;
    v2f a, bb;
    a[0] = X[m * 64 + kb];
    a[1] = X[m * 64 + kb + 1];
    bb[0] = cv ? RW[kb * 5 + cl] : 0.f;
    bb[1] = cv ? RW[(kb + 1) * 5 + cl] : 0.f;
    c = wmma_f32(a, bb, c);
  }
  int rloc = (l >> 4) << 3;
  float bc = cv ? lb[cl] : 0.f;
#pragma unroll
  for (int r = 0; r < 8; ++r) tile[(rloc + r) * 16 + cl] = c[r] + bc;
  __syncthreads();   // single-wave workgroup: orders LDS
  if (l < 16) {
    float v0 = tile[l * 16 + 0];
    float v1 = tile[l * 16 + 1];
    float v2 = tile[l * 16 + 2];
    float v3 = tile[l * 16 + 3];
    float v4 = tile[l * 16 + 4];
    float ss = v0 * v0 + v1 * v1 + v2 * v2 + v3 * v3 + v4 * v4;
    float s = 1.0f / fmaxf(sqrtf(ss), EPS_NORM);
    int rr = row0 + l;
    if (rr < n) {
      Y[rr * 5 + 0] = v0 * s;
      Y[rr * 5 + 1] = v1 * s;
      Y[rr * 5 + 2] = v2 * s;
      Y[rr * 5 + 3] = v3 * s;
      Y[rr * 5 + 4] = v4 * s;
    }
  }
}

// ============================ Host orchestration ============================
extern "C" void kernel_launch(void* const* d_in, const int* in_sizes, int n_in,
                              void* d_out, int out_size, void* d_ws, size_t ws_size,
                              hipStream_t stream) {
  const float* h     = (const float*)d_in[0];
  const int*   ei    = (const int*)d_in[1];
  const float* ew    = (const float*)d_in[2];
  const float* bg    = (const float*)d_in[3];
  const float* bbeta = (const float*)d_in[4];
  const float* c1_pw = (const float*)d_in[5];
  const float* c1_pb = (const float*)d_in[6];
  const float* c1_lw = (const float*)d_in[7];
  const float* c1_lb = (const float*)d_in[8];
  const float* c1_rw = (const float*)d_in[9];
  const float* c2_pw = (const float*)d_in[10];
  const float* c2_pb = (const float*)d_in[11];
  const float* c2_lw = (const float*)d_in[12];
  const float* c2_lb = (const float*)d_in[13];
  const float* c2_rw = (const float*)d_in[14];
  const float* c3_pw = (const float*)d_in[15];
  const float* c3_pb = (const float*)d_in[16];
  const float* c3_lw = (const float*)d_in[17];
  const float* c3_lb = (const float*)d_in[18];
  const float* c3_rw = (const float*)d_in[19];

  const int N = in_sizes[0] / 5;
  const int E = in_sizes[2];
  const int* src = ei;       // edge_index row 0
  const int* dst = ei + E;   // edge_index row 1

  // Workspace layout (floats)
  float* ws   = (float*)d_ws;
  float* mu   = ws;                          // 8 (5 used)
  float* rs   = ws + 8;                      // 8 (5 used)
  float* part = ws + 16;                     // 256 blocks * 10
  float* cnt  = part + 2560;                 // N
  float* invc = cnt + N;                     // N
  float* x0   = invc + N;                    // N*5 (BN output)
  float* xp1  = x0 + (size_t)N * 5;          // N*5
  float* agg1 = xp1 + (size_t)N * 5;         // N*5
  float* B1   = agg1 + (size_t)N * 5;        // N*64
  float* B2   = B1 + (size_t)N * 64;         // N*64
  float* B3   = B2 + (size_t)N * 64;         // N*64

  const int nb16 = (N + 15) / 16;

  // ---- BatchNorm (training-mode batch stats) ----
  bn_partial_k<<<256, 256, 0, stream>>>(h, part, N);
  bn_final_k<<<1, 32, 0, stream>>>(part, mu, rs, 256, 1.0f / (float)N);
  bn_apply_k<<<(N * 5 + 255) / 256, 256, 0, stream>>>(h, mu, rs, bg, bbeta, x0, N * 5);

  // ---- degrees (reused by all 3 layers) ----
  hipMemsetAsync(cnt, 0, (size_t)N * sizeof(float), stream);
  degree_k<<<(E + 255) / 256, 256, 0, stream>>>(dst, cnt, E);
  invcnt_k<<<(N + 255) / 256, 256, 0, stream>>>(cnt, invc, N);

  // ---- Layer 1: 5 -> 64 (scalar path, dims too small for WMMA) ----
  proj5_k<<<(N + 255) / 256, 256, 0, stream>>>(x0, c1_pw, c1_pb, xp1, N);
  hipMemsetAsync(agg1, 0, (size_t)N * 5 * sizeof(float), stream);
  edge_scatter5_k<<<(E + 255) / 256, 256, 0, stream>>>(src, dst, ew, xp1, agg1, E);
  node5to64_k<<<N, 64, 0, stream>>>(agg1, invc, x0, c1_lw, c1_lb, c1_rw, B1, N);

  // ---- Layer 2: 64 -> 64 (fp32 WMMA path) ----
  proj64_wmma_k<<<nb16, 128, 0, stream>>>(B1, c2_pw, c2_pb, B2, N);
  hipMemsetAsync(B3, 0, (size_t)N * 64 * sizeof(float), stream);
  edge_scatter64_k<<<(E * 16 + 255) / 256, 256, 0, stream>>>(src, dst, ew, B2, B3, E);
  node64_wmma_k<<<nb16, 128, 0, stream>>>(B3, invc, B1, c2_lw, c2_lb, c2_rw, B2, N);

  // ---- Layer 3: 64 -> 5 (fp32 WMMA, padded output tile) ----
  proj64_wmma_k<<<nb16, 128, 0, stream>>>(B2, c3_pw, c3_pb, B1, N);
  hipMemsetAsync(B3, 0, (size_t)N * 64 * sizeof(float), stream);
  edge_scatter64_k<<<(E * 16 + 255) / 256, 256, 0, stream>>>(src, dst, ew, B1, B3, E);
  node64to5_wmma_k<<<nb16, 32, 0, stream>>>(B3, invc, B2, c3_lw, c3_lb, c3_rw,
                                            (float*)d_out, N);
}